// MyGINConvV2_72086731096480
// MI455X (gfx1250) — compile-verified
//
#include <hip/hip_runtime.h>

#define NN 100000
#define NE 1000000
#define DD 64
#define NEG_SLOPE 0.01f

typedef __attribute__((ext_vector_type(2))) float v2f;
typedef __attribute__((ext_vector_type(8))) float v8f;

// ---------------------------------------------------------------------------
// Kernel 1: h = (1 + eps) * x   (written into d_out; float4 vectorized)
// ---------------------------------------------------------------------------
__global__ void gin_init(const float* __restrict__ x,
                         const float* __restrict__ eps,
                         float* __restrict__ h) {
  int i = blockIdx.x * blockDim.x + threadIdx.x;
  const int n4 = NN * DD / 4;
  if (i < n4) {
    float s = 1.0f + eps[0];
    float4 v = ((const float4*)x)[i];
    v.x *= s; v.y *= s; v.z *= s; v.w *= s;
    ((float4*)h)[i] = v;
  }
}

// ---------------------------------------------------------------------------
// Kernel 2: scatter-add  h[dst] += x[src].
// One wave (32 lanes) per edge; each lane handles 2 features (float2 load +
// 2 fp32 global atomics). Edge id forced into an SGPR via readfirstlane so
// the index loads become scalar s_load and the row base is SGPR-uniform.
// 125000 blocks * 8 waves = exactly 1M edges -> no guards, no divergence.
// edge_index is int32 (JAX x64 disabled): row 0 = src, row 1 = dst.
// ---------------------------------------------------------------------------
__global__ __launch_bounds__(256) void gin_scatter(const float* __restrict__ x,
                                                   const int* __restrict__ ei,
                                                   float* __restrict__ h) {
  const int e = __builtin_amdgcn_readfirstlane(blockIdx.x * 8 +
                                               (threadIdx.x >> 5));
  const unsigned lane = threadIdx.x & 31u;
  const int src = ei[e];        // uniform address -> scalar load
  const int dst = ei[NE + e];   // uniform address -> scalar load
  const float2 v = *(const float2*)(x + (size_t)src * DD + lane * 2u);
  float* p = h + (size_t)dst * DD + lane * 2u;
  unsafeAtomicAdd(p, v.x);
  unsafeAtomicAdd(p + 1, v.y);
}

// ---------------------------------------------------------------------------
// Kernel 3: in-place MLP: h = lrelu(lrelu(h@W1+b1)@W2+b2)
// One wave per 16-row block. fp32 WMMA 16x16x4. 64-thread blocks (2 waves),
// 3125 blocks * 2 waves * 16 rows = 100000 exactly -> zero divergence.
// ---------------------------------------------------------------------------
__global__ __launch_bounds__(64) void gin_mlp(float* __restrict__ h,
                                              const float* __restrict__ W1,
                                              const float* __restrict__ b1,
                                              const float* __restrict__ W2,
                                              const float* __restrict__ b2) {
  // LDS staging for h1, padded row stride (66 floats) to avoid bank conflicts
  __shared__ float sh[2][16][66];

  const unsigned lane = threadIdx.x & 31u;
  const unsigned wv   = threadIdx.x >> 5;     // wave in block: 0..1
  const unsigned lo   = lane & 15u;           // N (or M) index within tile
  const unsigned hi   = lane >> 4;            // K-pair selector (A/B layout)
  const int m0 = (blockIdx.x * 2 + wv) * 16;  // first row of this wave's block

  v8f zero = {};
  v8f c[4];
  #pragma unroll
  for (int n = 0; n < 4; ++n) c[n] = zero;

  // ---- layer 1: C = h[16x64] @ W1[64x64] ----
  #pragma unroll
  for (int kk = 0; kk < 16; ++kk) {
    const int k = 4 * kk + 2 * hi;
    // A (16x4 fp32): lane lo holds rows, hi selects K pair {k, k+1}
    v2f a = *(const v2f*)(h + (long)(m0 + lo) * DD + k);
    #pragma unroll
    for (int n = 0; n < 4; ++n) {
      v2f b;
      b.x = W1[k * DD + 16 * n + lo];
      b.y = W1[(k + 1) * DD + 16 * n + lo];
      c[n] = __builtin_amdgcn_wmma_f32_16x16x4_f32(
          false, a, false, b, (short)0, c[n], false, false);
    }
  }

  // bias + LeakyReLU -> LDS (C layout: VGPR r -> row r + 8*hi, col 16n+lo)
  #pragma unroll
  for (int n = 0; n < 4; ++n) {
    float bv = b1[16 * n + lo];
    #pragma unroll
    for (int r = 0; r < 8; ++r) {
      float v = c[n][r] + bv;
      v = (v >= 0.0f) ? v : NEG_SLOPE * v;
      sh[wv][r + 8 * hi][16 * n + lo] = v;
    }
  }
  __syncthreads();

  // ---- layer 2: C2 = h1[16x64] @ W2[64x64] ----
  v8f c2[4];
  #pragma unroll
  for (int n = 0; n < 4; ++n) c2[n] = zero;

  #pragma unroll
  for (int kk = 0; kk < 16; ++kk) {
    const int k = 4 * kk + 2 * hi;
    v2f a = *(const v2f*)&sh[wv][lo][k];
    #pragma unroll
    for (int n = 0; n < 4; ++n) {
      v2f b;
      b.x = W2[k * DD + 16 * n + lo];
      b.y = W2[(k + 1) * DD + 16 * n + lo];
      c2[n] = __builtin_amdgcn_wmma_f32_16x16x4_f32(
          false, a, false, b, (short)0, c2[n], false, false);
    }
  }

  // bias + LeakyReLU -> global (in place; this wave owns rows m0..m0+15)
  #pragma unroll
  for (int n = 0; n < 4; ++n) {
    float bv = b2[16 * n + lo];
    #pragma unroll
    for (int r = 0; r < 8; ++r) {
      float v = c2[n][r] + bv;
      v = (v >= 0.0f) ? v : NEG_SLOPE * v;
      h[(long)(m0 + r + 8 * hi) * DD + 16 * n + lo] = v;
    }
  }
}

// ---------------------------------------------------------------------------
extern "C" void kernel_launch(void* const* d_in, const int* in_sizes, int n_in,
                              void* d_out, int out_size, void* d_ws, size_t ws_size,
                              hipStream_t stream) {
  const float* x   = (const float*)d_in[0];
  const int*   ei  = (const int*)d_in[1];
  const float* eps = (const float*)d_in[2];
  const float* W1  = (const float*)d_in[3];
  const float* b1  = (const float*)d_in[4];
  const float* W2  = (const float*)d_in[5];
  const float* b2  = (const float*)d_in[6];
  float* out = (float*)d_out;

  // 1) h = (1+eps)*x  (into d_out)
  {
    int n4 = NN * DD / 4;
    int blocks = (n4 + 255) / 256;
    gin_init<<<blocks, 256, 0, stream>>>(x, eps, out);
  }
  // 2) scatter-add neighbor features (1 wave per edge)
  {
    int blocks = NE / 8;  // 8 waves of 32 per block, one edge per wave
    gin_scatter<<<blocks, 256, 0, stream>>>(x, ei, out);
  }
  // 3) fused 2-layer MLP with fp32 WMMA, in place on d_out
  {
    int blocks = NN / 32;  // 3125 blocks * 2 waves * 16 rows = 100000
    gin_mlp<<<blocks, 64, 0, stream>>>(out, W1, b1, W2, b2);
  }
}